// GATClassifierClassic_85272280695083
// MI455X (gfx1250) — compile-verified
//
#include <hip/hip_runtime.h>
#include <hip/hip_bf16.h>

typedef float v2f __attribute__((ext_vector_type(2)));
typedef float v8f __attribute__((ext_vector_type(8)));

#define F_IN 128
#define HC1  256      // H*C layer 1
#define H1   8
#define C1   32
#define OUTC 40
#define NEG_SLOPE 0.2f

// ---------------- float atomic max (CAS loop, handles negatives) ------------
__device__ __forceinline__ void atomicMaxFloat(float* addr, float val) {
    int* ai = (int*)addr;
    int old = __hip_atomic_load(ai, __ATOMIC_RELAXED, __HIP_MEMORY_SCOPE_AGENT);
    while (__int_as_float(old) < val) {
        int assumed = old;
        old = atomicCAS(ai, assumed, __float_as_int(val));
        if (old == assumed) break;
    }
}

// ---------------- generic fill ----------------------------------------------
__global__ void k_fill(float* __restrict__ p, float v, size_t n) {
    size_t i = (size_t)blockIdx.x * blockDim.x + threadIdx.x;
    if (i < n) p[i] = v;
}

// ---------------- WMMA GEMM: Y[N,ncols] = X[N,K] @ [WL;WR]^T + [bL;bR] ------
// One wave computes a 16 x (16*NB) strip: A fragment reused across NB WMMAs.
template <int NB>
__global__ void k_gemm_wmma(const float* __restrict__ X, int ldx,
                            const float* __restrict__ WL, const float* __restrict__ WR,
                            const float* __restrict__ bL, const float* __restrict__ bR,
                            int split, int ncols,
                            float* __restrict__ Y, int ldy, int K, int N) {
    int lane = threadIdx.x;                       // 0..31
    int ctg  = blockIdx.y * blockDim.y + threadIdx.y;   // column tile GROUP
    int col0 = ctg * 16 * NB;
    if (col0 >= ncols) return;                    // wave-uniform
    int row0 = blockIdx.x * 16;
    int l15  = lane & 15;
    int hi   = lane >> 4;                         // 0: K+0/1, 1: K+2/3

    int arow = row0 + l15; if (arow >= N) arow = N - 1;   // clamp, keep EXEC full
    const float* xrow = X + (size_t)arow * ldx;

    const float* wrow[NB];
    float bias[NB];
    #pragma unroll
    for (int b = 0; b < NB; ++b) {
        int col = col0 + 16 * b + l15;
        wrow[b] = (col < split) ? (WL + (size_t)col * K)
                                : (WR + (size_t)(col - split) * K);
        bias[b] = (col < split) ? bL[col] : bR[col - split];
    }

    v8f acc[NB];
    #pragma unroll
    for (int b = 0; b < NB; ++b) acc[b] = (v8f){0.f,0.f,0.f,0.f,0.f,0.f,0.f,0.f};

    for (int k = 0; k < K; k += 4) {
        int ka = k + hi * 2;                       // 8-byte aligned
        v2f a = *(const v2f*)(xrow + ka);           // A: 16x4, row-per-lane
        #pragma unroll
        for (int b = 0; b < NB; ++b) {
            v2f bb = *(const v2f*)(wrow[b] + ka);   // B: 4x16 = W^T column-per-lane
            acc[b] = __builtin_amdgcn_wmma_f32_16x16x4_f32(false, a, false, bb,
                                                           (short)0, acc[b], false, false);
        }
    }
    // D layout: vgpr i -> M = i + 8*hi, Ncol = col + 16*b
    #pragma unroll
    for (int b = 0; b < NB; ++b) {
        int col = col0 + 16 * b + l15;
        #pragma unroll
        for (int i = 0; i < 8; ++i) {
            int r = row0 + hi * 8 + i;
            if (r < N) Y[(size_t)r * ldy + col] = acc[b][i] + bias[b];
        }
    }
}

// ---------------- layer-1 edge logits: one wave per (edge, head) ------------
__global__ void k_logits1(const float* __restrict__ xlxr, const long long* __restrict__ ei,
                          int E, int E2, const float* __restrict__ att,
                          float* __restrict__ logits) {
    int e = blockIdx.x;
    int s, d;
    if (e < E) { s = (int)ei[e]; d = (int)ei[(size_t)E + e]; }
    else       { s = e - E;      d = e - E; }
    int h = threadIdx.y, c = threadIdx.x;
    float xl = xlxr[(size_t)s * 512 + h * C1 + c];
    float xr = xlxr[(size_t)d * 512 + HC1 + h * C1 + c];
    float v = xl + xr;
    v = (v > 0.f) ? v : NEG_SLOPE * v;
    v *= att[h * C1 + c];
    #pragma unroll
    for (int off = 16; off > 0; off >>= 1) v += __shfl_xor(v, off, 32);
    if (c == 0) logits[(size_t)e * H1 + h] = v;
}

// ---------------- segment max / denom over destinations ---------------------
__global__ void k_seg_max(const float* __restrict__ logits, const long long* __restrict__ ei,
                          int E, int E2, int Hh, float* __restrict__ m) {
    long long idx = (long long)blockIdx.x * blockDim.x + threadIdx.x;
    long long tot = (long long)E2 * Hh;
    if (idx >= tot) return;
    int e = (int)(idx / Hh), h = (int)(idx - (long long)e * Hh);
    int d = (e < E) ? (int)ei[(size_t)E + e] : (e - E);
    atomicMaxFloat(&m[(size_t)d * Hh + h], logits[idx]);
}

__global__ void k_seg_den(const float* __restrict__ logits, const long long* __restrict__ ei,
                          int E, int E2, int Hh, const float* __restrict__ m,
                          float* __restrict__ den) {
    long long idx = (long long)blockIdx.x * blockDim.x + threadIdx.x;
    long long tot = (long long)E2 * Hh;
    if (idx >= tot) return;
    int e = (int)(idx / Hh), h = (int)(idx - (long long)e * Hh);
    int d = (e < E) ? (int)ei[(size_t)E + e] : (e - E);
    atomicAdd(&den[(size_t)d * Hh + h], __expf(logits[idx] - m[(size_t)d * Hh + h]));
}

// ---------------- layer-1 weighted scatter: 256 lanes per edge --------------
__global__ void k_scatter1(const float* __restrict__ xlxr, const float* __restrict__ logits,
                           const float* __restrict__ m, const float* __restrict__ den,
                           const long long* __restrict__ ei, int E, int E2,
                           float* __restrict__ out1) {
    int e = blockIdx.x, t = threadIdx.x;                 // t = h*32+c
    int s, d;
    if (e < E) { s = (int)ei[e]; d = (int)ei[(size_t)E + e]; }
    else       { s = e - E;      d = e - E; }
    int h = t >> 5;
    float lg = logits[(size_t)e * H1 + h];
    float alpha = __expf(lg - m[(size_t)d * H1 + h]) / den[(size_t)d * H1 + h];
    float xl = xlxr[(size_t)s * 512 + t];
    atomicAdd(&out1[(size_t)d * HC1 + t], alpha * xl);
}

// ---------------- bias + ELU in place ---------------------------------------
__global__ void k_bias_elu(float* __restrict__ p, const float* __restrict__ b, size_t n, int ld) {
    size_t i = (size_t)blockIdx.x * blockDim.x + threadIdx.x;
    if (i >= n) return;
    float v = p[i] + b[i % ld];
    p[i] = (v > 0.f) ? v : (__expf(v) - 1.f);
}

// ---------------- layer-2 edge logits: one wave per edge --------------------
__global__ void k_logits2(const float* __restrict__ xlxr2, const long long* __restrict__ ei,
                          int E, int E2, const float* __restrict__ att,
                          float* __restrict__ logits) {
    int e = blockIdx.x * blockDim.y + threadIdx.y;
    if (e >= E2) return;
    int s, d;
    if (e < E) { s = (int)ei[e]; d = (int)ei[(size_t)E + e]; }
    else       { s = e - E;      d = e - E; }
    int c = threadIdx.x;
    float v = 0.f;
    for (int cc = c; cc < OUTC; cc += 32) {
        float xv = xlxr2[(size_t)s * 80 + cc] + xlxr2[(size_t)d * 80 + OUTC + cc];
        xv = (xv > 0.f) ? xv : NEG_SLOPE * xv;
        v += xv * att[cc];
    }
    #pragma unroll
    for (int off = 16; off > 0; off >>= 1) v += __shfl_xor(v, off, 32);
    if (c == 0) logits[e] = v;
}

// ---------------- layer-2 weighted scatter ----------------------------------
__global__ void k_scatter2(const float* __restrict__ xlxr2, const float* __restrict__ logits,
                           const float* __restrict__ m, const float* __restrict__ den,
                           const long long* __restrict__ ei, int E, int E2,
                           float* __restrict__ out2) {
    int e = blockIdx.x, t = threadIdx.x;
    if (t >= OUTC) return;
    int s, d;
    if (e < E) { s = (int)ei[e]; d = (int)ei[(size_t)E + e]; }
    else       { s = e - E;      d = e - E; }
    float alpha = __expf(logits[e] - m[d]) / den[d];
    atomicAdd(&out2[(size_t)d * OUTC + t], alpha * xlxr2[(size_t)s * 80 + t]);
}

// ---------------- per-node log_softmax: one wave per node -------------------
__global__ void k_logsoftmax(const float* __restrict__ out2, const float* __restrict__ b,
                             float* __restrict__ y, int N) {
    int n = blockIdx.x * blockDim.y + threadIdx.y;
    if (n >= N) return;
    int c = threadIdx.x;
    float v0 = out2[(size_t)n * OUTC + c] + b[c];                 // c in [0,32) < 40
    float v1 = (c < OUTC - 32) ? (out2[(size_t)n * OUTC + 32 + c] + b[32 + c]) : -3.0e38f;
    float mx = fmaxf(v0, v1);
    #pragma unroll
    for (int off = 16; off > 0; off >>= 1) mx = fmaxf(mx, __shfl_xor(mx, off, 32));
    float se = __expf(v0 - mx) + ((c < OUTC - 32) ? __expf(v1 - mx) : 0.f);
    #pragma unroll
    for (int off = 16; off > 0; off >>= 1) se += __shfl_xor(se, off, 32);
    float ls = mx + __logf(se);
    y[(size_t)n * OUTC + c] = v0 - ls;
    if (c < OUTC - 32) y[(size_t)n * OUTC + 32 + c] = v1 - ls;
}

extern "C" void kernel_launch(void* const* d_in, const int* in_sizes, int n_in,
                              void* d_out, int out_size, void* d_ws, size_t ws_size,
                              hipStream_t stream) {
    const float*     x    = (const float*)d_in[0];
    const long long* ei   = (const long long*)d_in[1];
    const float* Wl1 = (const float*)d_in[2];
    const float* bl1 = (const float*)d_in[3];
    const float* Wr1 = (const float*)d_in[4];
    const float* br1 = (const float*)d_in[5];
    const float* att1= (const float*)d_in[6];
    const float* b1  = (const float*)d_in[7];
    const float* Wl2 = (const float*)d_in[8];
    const float* bl2 = (const float*)d_in[9];
    const float* Wr2 = (const float*)d_in[10];
    const float* br2 = (const float*)d_in[11];
    const float* att2= (const float*)d_in[12];
    const float* b2  = (const float*)d_in[13];

    const int N  = in_sizes[0] / F_IN;
    const int E  = in_sizes[1] / 2;
    const int E2 = E + N;          // with self loops

    float* ws = (float*)d_ws;
    // layer-1 scratch
    size_t o_xlxr1 = 0;                                   // N*512
    size_t o_out1  = o_xlxr1 + (size_t)N * 512;           // N*256
    size_t o_log1  = o_out1  + (size_t)N * HC1;           // E2*8
    size_t o_m1    = o_log1  + (size_t)E2 * H1;           // N*8
    size_t o_d1    = o_m1    + (size_t)N * H1;            // N*8
    // layer-2 scratch reuses the (dead after scatter1) xlxr1 region
    size_t o_xlxr2 = 0;                                   // N*80
    size_t o_log2  = (size_t)N * 80;                      // E2
    size_t o_m2    = o_log2 + (size_t)E2;                 // N
    size_t o_d2    = o_m2   + (size_t)N;                  // N
    size_t o_out2  = o_d2   + (size_t)N;                  // N*40

    int rowTiles = (N + 15) / 16;

    // ---- layer 1 ----
    // xlxr1 = x @ [Wl1;Wr1]^T + [bl1;br1]
    // 512 cols = 8 groups of 16x64; block = 4 waves, grid.y = 2
    k_gemm_wmma<4><<<dim3(rowTiles, 2), dim3(32, 4), 0, stream>>>(
        x, F_IN, Wl1, Wr1, bl1, br1, HC1, 512, ws + o_xlxr1, 512, F_IN, N);

    k_fill<<<(int)(((size_t)N * H1 + 255) / 256), 256, 0, stream>>>(ws + o_m1, -3.0e38f, (size_t)N * H1);
    k_fill<<<(int)(((size_t)N * H1 + 255) / 256), 256, 0, stream>>>(ws + o_d1, 0.f, (size_t)N * H1);
    k_fill<<<(int)(((size_t)N * HC1 + 255) / 256), 256, 0, stream>>>(ws + o_out1, 0.f, (size_t)N * HC1);

    k_logits1<<<E2, dim3(32, 8), 0, stream>>>(ws + o_xlxr1, ei, E, E2, att1, ws + o_log1);

    int segBlocks = (int)(((size_t)E2 * H1 + 255) / 256);
    k_seg_max<<<segBlocks, 256, 0, stream>>>(ws + o_log1, ei, E, E2, H1, ws + o_m1);
    k_seg_den<<<segBlocks, 256, 0, stream>>>(ws + o_log1, ei, E, E2, H1, ws + o_m1, ws + o_d1);

    k_scatter1<<<E2, 256, 0, stream>>>(ws + o_xlxr1, ws + o_log1, ws + o_m1, ws + o_d1,
                                       ei, E, E2, ws + o_out1);

    k_bias_elu<<<(int)(((size_t)N * HC1 + 255) / 256), 256, 0, stream>>>(
        ws + o_out1, b1, (size_t)N * HC1, HC1);

    // ---- layer 2 ----
    // xlxr2 = h1 @ [Wl2;Wr2]^T + [bl2;br2]  (80 cols = one 16x80 strip per wave)
    k_gemm_wmma<5><<<dim3(rowTiles, 1), dim3(32, 1), 0, stream>>>(
        ws + o_out1, HC1, Wl2, Wr2, bl2, br2, OUTC, 80, ws + o_xlxr2, 80, HC1, N);

    k_fill<<<(N + 255) / 256, 256, 0, stream>>>(ws + o_m2, -3.0e38f, (size_t)N);
    k_fill<<<(N + 255) / 256, 256, 0, stream>>>(ws + o_d2, 0.f, (size_t)N);
    k_fill<<<(int)(((size_t)N * OUTC + 255) / 256), 256, 0, stream>>>(ws + o_out2, 0.f, (size_t)N * OUTC);

    k_logits2<<<(E2 + 3) / 4, dim3(32, 4), 0, stream>>>(ws + o_xlxr2, ei, E, E2, att2, ws + o_log2);

    int seg2 = (E2 + 255) / 256;
    k_seg_max<<<seg2, 256, 0, stream>>>(ws + o_log2, ei, E, E2, 1, ws + o_m2);
    k_seg_den<<<seg2, 256, 0, stream>>>(ws + o_log2, ei, E, E2, 1, ws + o_m2, ws + o_d2);

    k_scatter2<<<E2, 64, 0, stream>>>(ws + o_xlxr2, ws + o_log2, ws + o_m2, ws + o_d2,
                                      ei, E, E2, ws + o_out2);

    k_logsoftmax<<<(N + 3) / 4, dim3(32, 4), 0, stream>>>(ws + o_out2, b2, (float*)d_out, N);
}